// res_gnn_backbone_35880156791096
// MI455X (gfx1250) — compile-verified
//
#include <hip/hip_runtime.h>

#define NN   50000
#define EE   600000
#define DD   128
#define NBLK 2
#define EPSV 1e-5f
#define SLP  0.01f
#define LSTR 132   // LDS row stride in floats: bank-conflict-free for b64 reads

typedef float v2f __attribute__((ext_vector_type(2)));
typedef float v8f __attribute__((ext_vector_type(8)));

// ---------------- zero fill ----------------
__global__ __launch_bounds__(256)
void zero_kernel(float* __restrict__ p, int n) {
    int i = blockIdx.x * 256 + threadIdx.x;
    if (i < n) p[i] = 0.0f;
}

// ---------------- weighted SpMM scatter: Y[dst] += w * X[src] ----------------
// one wave (32 lanes) per edge, each lane handles 4 contiguous features (float4)
__global__ __launch_bounds__(256)
void spmm_scatter_kernel(const float* __restrict__ X,
                         const int* __restrict__ src,
                         const int* __restrict__ dst,
                         const float* __restrict__ ew,
                         float* __restrict__ Y) {
    int t = blockIdx.x * 256 + threadIdx.x;
    int e = t >> 5;
    int lane = t & 31;
    if (e >= EE) return;
    int s = src[e];
    int d = dst[e];
    float w = ew[e];
    float4 v = *(reinterpret_cast<const float4*>(X + (size_t)s * DD) + lane);
    float* out = Y + (size_t)d * DD + lane * 4;
    unsafeAtomicAdd(out + 0, w * v.x);
    unsafeAtomicAdd(out + 1, w * v.y);
    unsafeAtomicAdd(out + 2, w * v.z);
    unsafeAtomicAdd(out + 3, w * v.w);
}

// ---------------- fused TAGConv GEMM: H = X0*W0 + X1*W1 + X2*W2 + bias ----------------
// wave32 WMMA f32 16x16x4. Block = 8 waves; blockIdx.x = M tile (16 rows),
// wave id = N tile (16 cols). A strips staged in LDS once per block (shared
// by all 8 waves); B read from L2-hot weight matrices.
__global__ __launch_bounds__(256)
void tag_gemm_wmma_kernel(const float* __restrict__ X0,
                          const float* __restrict__ X1,
                          const float* __restrict__ X2,
                          const float* __restrict__ Wb,   // [3][DD][DD]
                          const float* __restrict__ bias, // [DD]
                          float* __restrict__ H) {
    __shared__ float lA[3 * 16 * LSTR];     // ~25 KB of the 320 KB WGP LDS
    const int t  = threadIdx.x;
    const int m0 = blockIdx.x * 16;

    const float* Xs[3] = {X0, X1, X2};
    // Stage A: 3 segments x 16 rows x 128 cols, coalesced b128 -> ds_store_b128
    for (int s = 0; s < 3; ++s) {
        const float* Xb = Xs[s] + (size_t)m0 * DD;
        float* Lb = lA + s * 16 * LSTR;
#pragma unroll
        for (int i = 0; i < 2; ++i) {
            int idx = t + i * 256;          // float4 index 0..511
            int row = idx >> 5;             // 0..15
            int col = (idx & 31) << 2;      // 0..124 step 4
            float4 v = *reinterpret_cast<const float4*>(Xb + (size_t)row * DD + col);
            *reinterpret_cast<float4*>(Lb + row * LSTR + col) = v;
        }
    }
    __syncthreads();

    const int lane = t & 31;
    const int wv   = t >> 5;                // 0..7 -> N tile
    const int n0   = wv * 16;
    const int kg   = (lane >> 4) * 2;       // lanes 0-15: K={0,1}; 16-31: K={2,3}
    const int mloc = lane & 15;             // A: lane holds row M = lane%16
    const int ncol = n0 + (lane & 15);      // B/C/D: lane holds col N = lane%16

    v8f acc = {};
    for (int sidx = 0; sidx < 3; ++sidx) {
        const float* Ap = lA + sidx * 16 * LSTR + mloc * LSTR + kg;   // LDS
        const float* Wp = Wb + (size_t)sidx * DD * DD + (size_t)kg * DD + ncol;
#pragma unroll
        for (int k = 0; k < DD; k += 4) {
            v2f a;                              // A 16x4 f32 per ISA layout (ds_load_b64)
            a.x = Ap[k];
            a.y = Ap[k + 1];
            v2f b;                              // B 4x16 f32 (rows kg, kg+1)
            b.x = Wp[(size_t)k * DD];
            b.y = Wp[(size_t)(k + 1) * DD];
            acc = __builtin_amdgcn_wmma_f32_16x16x4_f32(
                false, a, false, b, (short)0, acc, false, false);
        }
    }
    // C/D layout: VGPR r -> row m0 + r + 8*(lane/16), col = n0 + lane%16
    const float bv = bias[ncol];
    const int rowbase = m0 + ((lane >> 4) << 3);
#pragma unroll
    for (int r = 0; r < 8; ++r)
        H[(size_t)(rowbase + r) * DD + ncol] = acc[r] + bv;
}

// ---------------- BN column stats: stats[0:D]=sum, stats[D:2D]=sumsq ----------------
__global__ __launch_bounds__(128)
void bn_stats_kernel(const float* __restrict__ H, float* __restrict__ stats) {
    int col = threadIdx.x;                  // 0..127
    float s = 0.0f, sq = 0.0f;
    for (int r = blockIdx.x; r < NN; r += gridDim.x) {
        float v = H[(size_t)r * DD + col];
        s  += v;
        sq += v * v;
    }
    unsafeAtomicAdd(&stats[col], s);
    unsafeAtomicAdd(&stats[DD + col], sq);
}

// ---------------- BN apply + leaky_relu + residual: Y += act(norm(H)) ----------------
__global__ __launch_bounds__(256)
void bn_apply_kernel(const float* __restrict__ H,
                     const float* __restrict__ stats,
                     const float* __restrict__ gamma,
                     const float* __restrict__ beta,
                     float* __restrict__ Y) {
    int idx = blockIdx.x * 256 + threadIdx.x;
    if (idx >= NN * DD) return;
    int col = idx & (DD - 1);
    const float invn = 1.0f / (float)NN;
    float mean = stats[col] * invn;
    float var  = stats[DD + col] * invn - mean * mean;
    float scale = gamma[col] * rsqrtf(var + EPSV);
    float hv = (H[idx] - mean) * scale + beta[col];
    hv = (hv >= 0.0f) ? hv : SLP * hv;
    Y[idx] += hv;
}

extern "C" void kernel_launch(void* const* d_in, const int* in_sizes, int n_in,
                              void* d_out, int out_size, void* d_ws, size_t ws_size,
                              hipStream_t stream) {
    const float* y  = (const float*)d_in[0];   // [N, D]
    const int*   ei = (const int*)d_in[1];     // [2, E]
    const float* ew = (const float*)d_in[2];   // [E]
    const float* lw = (const float*)d_in[3];   // [NB, 3, D, D]
    const float* cb = (const float*)d_in[4];   // [NB, D]
    const float* gm = (const float*)d_in[5];   // [NB, D]
    const float* bt = (const float*)d_in[6];   // [NB, D]
    float* Y = (float*)d_out;

    const size_t nd = (size_t)NN * DD;
    float* x1    = (float*)d_ws;
    float* x2    = x1 + nd;
    float* h     = x2 + nd;
    float* stats = h + nd;                     // 2*DD floats

    const int* src = ei;
    const int* dst = ei + EE;

    // running y lives in d_out
    hipMemcpyAsync(Y, y, nd * sizeof(float), hipMemcpyDeviceToDevice, stream);

    const int zgrid2 = (int)((2 * nd + 255) / 256);
    const int zgridS = 1;                      // 256 >= 2*DD
    const int sgrid  = (EE * 32) / 256;        // 75000 blocks, one wave per edge
    const int ggrid  = NN / 16;                // 3125 M tiles
    const int agrid  = (int)((nd + 255) / 256);

    for (int b = 0; b < NBLK; ++b) {
        zero_kernel<<<zgrid2, 256, 0, stream>>>(x1, (int)(2 * nd));   // x1 & x2
        zero_kernel<<<zgridS, 256, 0, stream>>>(stats, 2 * DD);
        spmm_scatter_kernel<<<sgrid, 256, 0, stream>>>(Y,  src, dst, ew, x1);
        spmm_scatter_kernel<<<sgrid, 256, 0, stream>>>(x1, src, dst, ew, x2);
        tag_gemm_wmma_kernel<<<ggrid, 256, 0, stream>>>(
            Y, x1, x2, lw + (size_t)b * 3 * DD * DD, cb + (size_t)b * DD, h);
        bn_stats_kernel<<<256, 128, 0, stream>>>(h, stats);
        bn_apply_kernel<<<agrid, 256, 0, stream>>>(
            h, stats, gm + (size_t)b * DD, bt + (size_t)b * DD, Y);
    }
}